// CompoundLoss_65910568124906
// MI455X (gfx1250) — compile-verified
//
#include <hip/hip_runtime.h>
#include <math.h>

// ---------------------------------------------------------------------------
// CompoundLoss for MI455X (gfx1250, wave32).
// Dominant cost: Gram matrix X*X^T (4096x4096x512 fp32 = 17.2 GFLOP) for
// batch-hard triplet mining -> done with V_WMMA_F32_16X16X4_F32 tiles,
// reduced on the fly (distance matrix never materialized).
// ---------------------------------------------------------------------------

typedef float v2f __attribute__((ext_vector_type(2)));
typedef float v8f __attribute__((ext_vector_type(8)));

#define F_ALPHA   1.0f
#define F_MARGIN  0.3f
#define KD_T      3.0f
#define NEG_FILL  1000000.0f

// ---------------------------------------------------------------------------
// Kernel 1: row squared-norms + init of mining buffers. One wave32 per row.
// ---------------------------------------------------------------------------
__global__ void __launch_bounds__(256)
k_norms(const float* __restrict__ feat, float* __restrict__ sq,
        float* __restrict__ hp, float* __restrict__ hn, int B, int D)
{
    int gtid = blockIdx.x * blockDim.x + threadIdx.x;
    int wave = gtid >> 5;
    int lane = gtid & 31;
    if (wave >= B) return;
    const float* row = feat + (size_t)wave * D;
    float s = 0.0f;
    for (int k = lane; k < D; k += 32) {
        float v = row[k];
        s = fmaf(v, v, s);
    }
    #pragma unroll
    for (int m = 16; m >= 1; m >>= 1)
        s += __shfl_xor(s, m, 32);
    if (lane == 0) {
        sq[wave] = s;
        hp[wave] = 0.0f;       // hardest positive: max of non-negative values
        hn[wave] = NEG_FILL;   // hardest negative: min, fill = 1e6
    }
}

// ---------------------------------------------------------------------------
// Kernel 2: tiled X*X^T with fp32 WMMA + on-the-fly batch-hard mining.
// 256 threads = 8 waves per block; block computes a 64x64 tile of distances.
// Wave w owns subtile row (w&3) and two subtile cols (w>>2)*2 .. +1.
// A-operand layout (ISA 7.12.2, 32-bit A 16x4): lane%16 = M, VGPRv/half: K=v+2*half.
// C/D layout: lane%16 = N, VGPRv/half: M = v + 8*half.
// ---------------------------------------------------------------------------
__global__ void __launch_bounds__(256)
k_gram(const float* __restrict__ feat, const float* __restrict__ sq,
       const int* __restrict__ labels,
       float* __restrict__ hp, float* __restrict__ hn, int B, int D)
{
    __shared__ float As[64][33];   // pad to 33 floats: conflict-free column walks
    __shared__ float Bs[64][33];
    __shared__ int   La[64], Lb[64];
    __shared__ float Sqa[64], Sqb[64];

    const int I    = blockIdx.x * 64;
    const int J    = blockIdx.y * 64;
    const int tid  = threadIdx.x;
    const int lane = tid & 31;
    const int w    = tid >> 5;
    const int mi   = w & 3;            // subtile row index (0..3)
    const int nb   = (w >> 2) * 2;     // subtile col base (0 or 2), 2 tiles/wave
    const int half = lane >> 4;        // 0/1
    const int l16  = lane & 15;

    if (tid < 64) {
        La[tid]  = labels[I + tid];
        Lb[tid]  = labels[J + tid];
        Sqa[tid] = sq[I + tid];
        Sqb[tid] = sq[J + tid];
    }

    v8f acc0 = {0.f,0.f,0.f,0.f,0.f,0.f,0.f,0.f};
    v8f acc1 = {0.f,0.f,0.f,0.f,0.f,0.f,0.f,0.f};

    for (int kb = 0; kb < D; kb += 32) {
        __syncthreads();
        // cooperative load: 64x32 panel each, 8 floats/thread, coalesced
        #pragma unroll
        for (int r = 0; r < 8; ++r) {
            int idx = tid + r * 256;
            int row = idx >> 5, col = idx & 31;
            As[row][col] = feat[(size_t)(I + row) * D + kb + col];
            Bs[row][col] = feat[(size_t)(J + row) * D + kb + col];
        }
        __syncthreads();

        #pragma unroll
        for (int k0 = 0; k0 < 32; k0 += 4) {
            int ka = k0 + 2 * half;    // K index for this half / VGPR pair
            v2f a, b0, b1;
            a.x  = As[mi * 16 + l16][ka];
            a.y  = As[mi * 16 + l16][ka + 1];
            b0.x = Bs[nb * 16 + l16][ka];
            b0.y = Bs[nb * 16 + l16][ka + 1];
            b1.x = Bs[(nb + 1) * 16 + l16][ka];
            b1.y = Bs[(nb + 1) * 16 + l16][ka + 1];
            acc0 = __builtin_amdgcn_wmma_f32_16x16x4_f32(
                       false, a, false, b0, (short)0, acc0, false, false);
            acc1 = __builtin_amdgcn_wmma_f32_16x16x4_f32(
                       false, a, false, b1, (short)0, acc1, false, false);
        }
    }

    // Epilogue: gram -> distance -> masked row-wise max/min -> global atomics.
    #pragma unroll
    for (int t = 0; t < 2; ++t) {
        v8f acc = (t == 0) ? acc0 : acc1;
        int jn  = nb + t;
        int j   = J + jn * 16 + l16;           // global column for this lane
        int lb  = Lb[jn * 16 + l16];
        float sqj = Sqb[jn * 16 + l16];
        #pragma unroll
        for (int v = 0; v < 8; ++v) {
            int m = v + 8 * half;
            int i = I + mi * 16 + m;           // global row
            float d2 = Sqa[mi * 16 + m] + sqj - 2.0f * acc[v];
            d2 = fmaxf(d2, 0.0f);
            float d = sqrtf(d2);
            bool same = (La[mi * 16 + m] == lb);
            float pd = (same && (i != j)) ? d : 0.0f;       // positives (diag excluded)
            float nd = same ? NEG_FILL : d;                 // negatives (fill elsewhere)
            // reduce across the 16 lanes of this half (same row i, different j)
            #pragma unroll
            for (int msk = 8; msk >= 1; msk >>= 1) {
                pd = fmaxf(pd, __shfl_xor(pd, msk, 32));
                nd = fminf(nd, __shfl_xor(nd, msk, 32));
            }
            if (l16 == 0) {
                // non-negative floats: int bit-pattern order == float order
                atomicMax((int*)&hp[i], __float_as_int(pd));
                atomicMin((int*)&hn[i], __float_as_int(nd));
            }
        }
    }
}

// ---------------------------------------------------------------------------
// Kernel 3: focal + KD + triplet combine. Single 256-thread block.
// ---------------------------------------------------------------------------
__global__ void __launch_bounds__(256)
k_final(const float* __restrict__ logits, const int* __restrict__ labels,
        const float* __restrict__ soft,
        const float* __restrict__ hp, const float* __restrict__ hn,
        float* __restrict__ out, int B)
{
    __shared__ float rf[256], rk[256], rt[256], rv[256];
    int tid = threadIdx.x;
    float fsum = 0.f, ksum = 0.f, tsum = 0.f, vsum = 0.f;

    for (int i = tid; i < B; i += 256) {
        float l0 = logits[2 * i], l1 = logits[2 * i + 1];
        int   y  = labels[i];
        // focal (gamma = 2)
        float mx  = fmaxf(l0, l1);
        float lse = mx + logf(expf(l0 - mx) + expf(l1 - mx));
        float ce  = lse - (y ? l1 : l0);
        float pt  = expf(-ce);
        float om  = 1.0f - pt;
        fsum += F_ALPHA * om * om * ce;
        // KD: KL(teacher || student_log) * T^2, batchmean
        float s0 = l0 / KD_T, s1 = l1 / KD_T;
        float ms = fmaxf(s0, s1);
        float lss = ms + logf(expf(s0 - ms) + expf(s1 - ms));
        float sl0 = s0 - lss, sl1 = s1 - lss;
        float t0 = logf(soft[2 * i] + 1e-8f) / KD_T;
        float t1 = logf(soft[2 * i + 1] + 1e-8f) / KD_T;
        float mt = fmaxf(t0, t1);
        float e0 = expf(t0 - mt), e1 = expf(t1 - mt);
        float Z  = e0 + e1, lZ = logf(Z);
        float te0 = e0 / Z, te1 = e1 / Z;
        float lt0 = t0 - mt - lZ, lt1 = t1 - mt - lZ;
        ksum += te0 * (lt0 - sl0) + te1 * (lt1 - sl1);
        // triplet per-row (valid <=> max positive distance > 0 <=> sum > 0)
        float p = hp[i], nmin = hn[i];
        float per   = fmaxf(p - nmin + F_MARGIN, 0.0f);
        float valid = (p > 0.0f) ? 1.0f : 0.0f;
        tsum += per * valid;
        vsum += valid;
    }

    rf[tid] = fsum; rk[tid] = ksum; rt[tid] = tsum; rv[tid] = vsum;
    __syncthreads();
    for (int s = 128; s > 0; s >>= 1) {
        if (tid < s) {
            rf[tid] += rf[tid + s];
            rk[tid] += rk[tid + s];
            rt[tid] += rt[tid + s];
            rv[tid] += rv[tid + s];
        }
        __syncthreads();
    }
    if (tid == 0) {
        float focal = rf[0] / (float)B;
        float kd    = rk[0] / (float)B * (KD_T * KD_T);
        float trip  = (rv[0] > 0.0f) ? rt[0] / fmaxf(rv[0], 1.0f) : 0.0f;
        out[0] = 1.0f * focal + 0.5f * trip + 1.0f * kd;  // total
        out[1] = focal;
        out[2] = trip;
        out[3] = kd;
    }
}

// ---------------------------------------------------------------------------
extern "C" void kernel_launch(void* const* d_in, const int* in_sizes, int n_in,
                              void* d_out, int out_size, void* d_ws, size_t ws_size,
                              hipStream_t stream)
{
    const float* logits = (const float*)d_in[0];   // [B, 2]
    const float* feat   = (const float*)d_in[1];   // [B, D]
    const int*   labels = (const int*)d_in[2];     // [B]
    const float* soft   = (const float*)d_in[3];   // [B, 2]

    const int B = in_sizes[2];            // 4096
    const int D = in_sizes[1] / B;        // 512

    float* sq = (float*)d_ws;             // [B]
    float* hp = sq + B;                   // [B] hardest positive (max)
    float* hn = hp + B;                   // [B] hardest negative (min)

    // 1) norms + init (one wave per row, 8 waves per block)
    int nblk = (B + 7) / 8;
    k_norms<<<nblk, 256, 0, stream>>>(feat, sq, hp, hn, B, D);

    // 2) WMMA gram + mining: 64x64 distance tile per workgroup
    dim3 grid(B / 64, B / 64);
    k_gram<<<grid, 256, 0, stream>>>(feat, sq, labels, hp, hn, B, D);

    // 3) combine all three losses
    k_final<<<1, 256, 0, stream>>>(logits, labels, soft, hp, hn, (float*)d_out, B);
}